// GATE_module_11888469476241
// MI455X (gfx1250) — compile-verified
//
#include <hip/hip_runtime.h>
#include <math.h>

#define NN 100000
#define EE 3200000
#define CIN 15
#define CE 2
#define HID 20
#define HEADS 4
#define DH 5
#define NG 10
#define PER (NN / NG)
#define FCC 15
#define NCONV 5
#define NUMFC 5

typedef __attribute__((ext_vector_type(2))) float v2f;
typedef __attribute__((ext_vector_type(8))) float v8f;

__device__ __forceinline__ float fixnum(float v) {
  float t = (v != v) ? 0.f : v;  // NaN -> 0 (branchless select)
  return fminf(fmaxf(t, -3.4028234663852886e38f), 3.4028234663852886e38f);
}

// order-preserving monotone encoding of f32 into u32 for atomicMax
__device__ __forceinline__ unsigned fenc(float f) {
  unsigned u = __float_as_uint(f);
  return (u & 0x80000000u) ? ~u : (u | 0x80000000u);
}
__device__ __forceinline__ float fdec(unsigned u) {
  unsigned v = (u & 0x80000000u) ? (u & 0x7fffffffu) : ~u;
  return __uint_as_float(v);
}
#define ENC_NEG_INF 0x007fffffu  // fenc(-inf)

// ---------------------------------------------------------------------------
// Tiled node GEMM on V_WMMA_F32_16X16X4_F32.
// One wave owns one 16-wide column tile: it caches the B fragments (W) and the
// fused BatchNorm scale/shift in registers ONCE, then streams row tiles.
// All guards are branchless (clamped indices + selects); requires nrows%16==0.
// WPC = waves per column tile (row-tile stride).
// ---------------------------------------------------------------------------
template <int K, int OUT, int WPC, bool BN, bool NANFIX, bool RESID, bool PE>
__global__ __launch_bounds__(128) void gemm_node(
    const float* __restrict__ in, const float* __restrict__ W,
    const float* __restrict__ bias, const float* __restrict__ gamma,
    const float* __restrict__ beta, const float* __restrict__ mean,
    const float* __restrict__ var, const float* __restrict__ resid,
    float* __restrict__ out, int nrows) {
  constexpr int COLT = (OUT + 15) / 16;
  constexpr int KS = (K + 3) / 4;
  const int rowTiles = nrows >> 4;  // nrows % 16 == 0 by construction
  int wave = blockIdx.x * (blockDim.x >> 5) + (threadIdx.x >> 5);
  int ct = wave % COLT;
  int rt0 = wave / COLT;
  int lane = threadIdx.x & 31;
  int r = lane & 15;
  int khalf = (lane >> 4) << 1;  // A/B: lanes 0-15 hold K+0/K+1, 16-31 hold K+2/K+3

  int cB = ct * 16 + r;
  int cBc = cB < OUT ? cB : OUT - 1;  // clamped (safe) column
  bool cOK = cB < OUT;

  // ---- per-wave setup: B fragments + fused BN constants (registers) ----
  v2f bfrag[KS];
  float asc[KS][2], ash[KS][2];
#pragma unroll
  for (int s = 0; s < KS; ++s) {
#pragma unroll
    for (int i = 0; i < 2; ++i) {
      int k = s * 4 + khalf + i;
      int kc = k < K ? k : K - 1;  // clamped (safe) k
      float bv = W[kc * OUT + cBc];
      bv = (k < K && cOK) ? bv : 0.f;  // zero B => garbage A is harmless
      if (i == 0) bfrag[s].x = bv; else bfrag[s].y = bv;
      if (BN) {
        float sc = rsqrtf(var[kc] + 1e-5f) * gamma[kc];
        asc[s][i] = sc;
        ash[s][i] = beta[kc] - mean[kc] * sc;
      }
    }
  }
  float bb = bias ? bias[cBc] : 0.f;
  float dexp = 0.f;
  if (PE) dexp = expf(-0.9210340371976184f * (float)(cB >> 1));  // 10000^{-j/10}

  // ---- stream row tiles ----
  for (int rt = rt0; rt < rowTiles; rt += WPC) {
    const float* inrow = in + (rt * 16 + r) * K;
    v8f acc = {0.f, 0.f, 0.f, 0.f, 0.f, 0.f, 0.f, 0.f};
#pragma unroll
    for (int s = 0; s < KS; ++s) {
      v2f a;
#pragma unroll
      for (int i = 0; i < 2; ++i) {
        int k = s * 4 + khalf + i;
        int kc = k < K ? k : K - 1;
        float av = inrow[kc];
        if (NANFIX) av = fixnum(av);
        if (BN) av = fmaxf(av * asc[s][i] + ash[s][i], 0.f);
        if (i == 0) a.x = av; else a.y = av;
      }
      acc = __builtin_amdgcn_wmma_f32_16x16x4_f32(false, a, false, bfrag[s],
                                                  (short)0, acc, false, false);
    }
    // D layout: lanes 0-15 -> M=v, N=lane; lanes 16-31 -> M=v+8, N=lane-16
    if (cOK) {
      int rbase = rt * 16 + ((lane >> 4) << 3);
#pragma unroll
      for (int v = 0; v < 8; ++v) {
        int row = rbase + v;
        float val = acc[v] + bb;
        if (RESID) val += resid[row * OUT + cB];
        if (PE) {
          float ang = (float)(row % PER) * dexp;
          val += (cB & 1) ? cosf(ang) : sinf(ang);
        }
        out[row * OUT + cB] = val;
      }
    }
  }
}

// ---------------------------------------------------------------------------
__global__ void fix_edges(const float* __restrict__ in, float* __restrict__ out,
                          int n) {
  int i = blockIdx.x * blockDim.x + threadIdx.x;
  if (i < n) out[i] = fixnum(in[i]);
}

// Ae[c*HEADS+h] = sum_d We[c, h*DH+d] * a_e[h,d]   (folds ep projection)
__global__ void ae_precompute(const float* __restrict__ We,
                              const float* __restrict__ a_e,
                              float* __restrict__ Ae) {
  int t = threadIdx.x;
  if (t < CE * HEADS) {
    int c = t / HEADS, h = t % HEADS;
    float s = 0.f;
    for (int d = 0; d < DH; ++d) s += We[c * HID + h * DH + d] * a_e[h * DH + d];
    Ae[t] = s;
  }
}

// per-node: attention dot products + reset of softmax/agg state
__global__ void node_init(const float* __restrict__ xp,
                          const float* __restrict__ a_src,
                          const float* __restrict__ a_dst,
                          float* __restrict__ ssrc, float* __restrict__ sdst,
                          unsigned* __restrict__ menc, float* __restrict__ den,
                          float* __restrict__ agg) {
  int n = blockIdx.x * blockDim.x + threadIdx.x;
  if (n >= NN) return;
  float xr[HID];
  const float4* p = (const float4*)(xp + n * HID);
#pragma unroll
  for (int q = 0; q < 5; ++q) {
    float4 v = p[q];
    xr[q * 4] = v.x; xr[q * 4 + 1] = v.y; xr[q * 4 + 2] = v.z; xr[q * 4 + 3] = v.w;
  }
#pragma unroll
  for (int h = 0; h < HEADS; ++h) {
    float ss = 0.f, sd = 0.f;
#pragma unroll
    for (int d = 0; d < DH; ++d) {
      float v = xr[h * DH + d];
      ss += v * a_src[h * DH + d];
      sd += v * a_dst[h * DH + d];
    }
    ssrc[n * HEADS + h] = ss;
    sdst[n * HEADS + h] = sd;
    menc[n * HEADS + h] = ENC_NEG_INF;
    den[n * HEADS + h] = 0.f;
  }
#pragma unroll
  for (int c = 0; c < HID; ++c) agg[n * HID + c] = 0.f;
}

__global__ void edge_logit(const int* __restrict__ ei,
                           const float* __restrict__ e_in,
                           const float* __restrict__ ssrc,
                           const float* __restrict__ sdst,
                           const float* __restrict__ Ae,
                           float* __restrict__ logit,
                           unsigned* __restrict__ menc) {
  int e = blockIdx.x * blockDim.x + threadIdx.x;
  if (e >= EE) return;
  int s = ei[e], d = ei[EE + e];
  float e0 = e_in[e * 2], e1 = e_in[e * 2 + 1];
#pragma unroll
  for (int h = 0; h < HEADS; ++h) {
    float l = sdst[d * HEADS + h] + ssrc[s * HEADS + h] + e0 * Ae[h] +
              e1 * Ae[HEADS + h];
    l = l > 0.f ? l : 0.2f * l;  // leaky_relu(0.2)
    logit[e * HEADS + h] = l;
    atomicMax(&menc[d * HEADS + h], fenc(l));
  }
}

// accumulate softmax denominators (ex recomputed later; saves a 51MB stream)
__global__ void edge_exp(const int* __restrict__ ei,
                         const float* __restrict__ logit,
                         const unsigned* __restrict__ menc,
                         float* __restrict__ den) {
  int e = blockIdx.x * blockDim.x + threadIdx.x;
  if (e >= EE) return;
  int d = ei[EE + e];
#pragma unroll
  for (int h = 0; h < HEADS; ++h) {
    float m = fdec(menc[d * HEADS + h]);
    float ex = expf(logit[e * HEADS + h] - m);
    atomicAdd(&den[d * HEADS + h], ex);
  }
}

__global__ void edge_agg(const int* __restrict__ ei,
                         const float* __restrict__ logit,
                         const unsigned* __restrict__ menc,
                         const float* __restrict__ den,
                         const float* __restrict__ xp, float* __restrict__ agg) {
  int e = blockIdx.x * blockDim.x + threadIdx.x;
  if (e >= EE) return;
  int s = ei[e], d = ei[EE + e];
  float al[HEADS];
#pragma unroll
  for (int h = 0; h < HEADS; ++h) {
    float m = fdec(menc[d * HEADS + h]);
    float ex = expf(logit[e * HEADS + h] - m);
    al[h] = ex / (den[d * HEADS + h] + 1e-16f);
  }
  const float4* p = (const float4*)(xp + s * HID);
#pragma unroll
  for (int q = 0; q < 5; ++q) {
    float4 v = p[q];
    int b = q * 4;
    atomicAdd(&agg[d * HID + b + 0], al[(b + 0) / DH] * v.x);
    atomicAdd(&agg[d * HID + b + 1], al[(b + 1) / DH] * v.y);
    atomicAdd(&agg[d * HID + b + 2], al[(b + 2) / DH] * v.z);
    atomicAdd(&agg[d * HID + b + 3], al[(b + 3) / DH] * v.w);
  }
}

__global__ void node_update(const float* __restrict__ x,
                            const float* __restrict__ agg,
                            const float* __restrict__ bx,
                            float* __restrict__ xn) {
  int i = blockIdx.x * blockDim.x + threadIdx.x;
  if (i >= NN * HID) return;
  int c = i % HID;
  xn[i] = x[i] + fmaxf(agg[i] + bx[c], 0.f);
}

// en = e + relu(concat(xn[src], xn[dst], e) @ Wem + bem); Wem cached in LDS
__global__ void edge_mlp(const float* __restrict__ xn,
                         const float* __restrict__ e_in,
                         const int* __restrict__ ei,
                         const float* __restrict__ Wem,
                         const float* __restrict__ bem,
                         float* __restrict__ e_out) {
  __shared__ float sW[86];  // 84 Wem + 2 bem
  int t = threadIdx.x;
  if (t < 86) sW[t] = (t < 84) ? Wem[t] : bem[t - 84];
  __syncthreads();
  int e = blockIdx.x * blockDim.x + t;
  if (e >= EE) return;
  int s = ei[e], d = ei[EE + e];
  float a0 = sW[84], a1 = sW[85];
  const float4* xs = (const float4*)(xn + s * HID);
  const float4* xd = (const float4*)(xn + d * HID);
#pragma unroll
  for (int q = 0; q < 5; ++q) {
    float4 v = xs[q];
    int k = q * 4;
    a0 += v.x * sW[(k + 0) * 2] + v.y * sW[(k + 1) * 2] + v.z * sW[(k + 2) * 2] +
          v.w * sW[(k + 3) * 2];
    a1 += v.x * sW[(k + 0) * 2 + 1] + v.y * sW[(k + 1) * 2 + 1] +
          v.z * sW[(k + 2) * 2 + 1] + v.w * sW[(k + 3) * 2 + 1];
  }
#pragma unroll
  for (int q = 0; q < 5; ++q) {
    float4 v = xd[q];
    int k = HID + q * 4;
    a0 += v.x * sW[(k + 0) * 2] + v.y * sW[(k + 1) * 2] + v.z * sW[(k + 2) * 2] +
          v.w * sW[(k + 3) * 2];
    a1 += v.x * sW[(k + 0) * 2 + 1] + v.y * sW[(k + 1) * 2 + 1] +
          v.z * sW[(k + 2) * 2 + 1] + v.w * sW[(k + 3) * 2 + 1];
  }
  float e0 = e_in[e * 2], e1 = e_in[e * 2 + 1];
  a0 += e0 * sW[80] + e1 * sW[82];
  a1 += e0 * sW[81] + e1 * sW[83];
  e_out[e * 2] = e0 + fmaxf(a0, 0.f);
  e_out[e * 2 + 1] = e1 + fmaxf(a1, 0.f);
}

// ---------------------------------------------------------------------------
extern "C" void kernel_launch(void* const* d_in, const int* in_sizes, int n_in,
                              void* d_out, int out_size, void* d_ws,
                              size_t ws_size, hipStream_t stream) {
  (void)n_in; (void)out_size; (void)ws_size;
  struct Lin { const float *gamma, *beta, *mean, *var, *W, *b; };
  struct Conv { const float *Wx, *We, *a_src, *a_dst, *a_e, *bx, *Wem, *bem; };
  const float* X; const float* EA; const int* EI;
  Lin emb[2], fc[NUMFC]; Conv cv[2];
  auto F = [&](int i) { return (const float*)d_in[i]; };

  if (in_sizes[0] == NN * CIN) {
    // insertion-order recursive flatten: x, edge_attr, params{emb,conv0,conv_sh,fc}, edge_index, batch_idx
    X = F(0); EA = F(1);
    int i = 2;
    for (int m = 0; m < 2; ++m) { emb[m] = {F(i), F(i + 1), F(i + 2), F(i + 3), F(i + 4), F(i + 5)}; i += 6; }
    for (int m = 0; m < 2; ++m) { cv[m] = {F(i), F(i + 1), F(i + 2), F(i + 3), F(i + 4), F(i + 5), F(i + 6), F(i + 7)}; i += 8; }
    for (int m = 0; m < NUMFC; ++m) { fc[m] = {F(i), F(i + 1), F(i + 2), F(i + 3), F(i + 4), F(i + 5)}; i += 6; }
    EI = (const int*)d_in[i];
  } else {
    // jax sorted-key flatten: batch_idx, edge_attr, edge_index, params{conv0,conv_sh,emb,fc}, x
    EA = F(1); EI = (const int*)d_in[2];
    int i = 3;
    for (int m = 0; m < 2; ++m) {  // conv keys sorted: We,Wem,Wx,a_dst,a_e,a_src,bem,bx
      cv[m].We = F(i); cv[m].Wem = F(i + 1); cv[m].Wx = F(i + 2);
      cv[m].a_dst = F(i + 3); cv[m].a_e = F(i + 4); cv[m].a_src = F(i + 5);
      cv[m].bem = F(i + 6); cv[m].bx = F(i + 7); i += 8;
    }
    for (int m = 0; m < 2; ++m) {  // lin keys sorted: W,b,beta,gamma,mean,var
      emb[m].W = F(i); emb[m].b = F(i + 1); emb[m].beta = F(i + 2);
      emb[m].gamma = F(i + 3); emb[m].mean = F(i + 4); emb[m].var = F(i + 5); i += 6;
    }
    for (int m = 0; m < NUMFC; ++m) {
      fc[m].W = F(i); fc[m].b = F(i + 1); fc[m].beta = F(i + 2);
      fc[m].gamma = F(i + 3); fc[m].mean = F(i + 4); fc[m].var = F(i + 5); i += 6;
    }
    X = F(i);
  }

  // workspace carving (256B-aligned regions)
  size_t off = 0;
  auto take = [&](size_t nfl) {
    void* p = (char*)d_ws + off;
    off += ((nfl * sizeof(float) + 255) & ~(size_t)255);
    return p;
  };
  float* xa = (float*)take((size_t)NN * HID);
  float* xb = (float*)take((size_t)NN * HID);
  float* ea = (float*)take((size_t)EE * CE);
  float* eb = (float*)take((size_t)EE * CE);
  float* xp = (float*)take((size_t)NN * HID);
  float* ssrc = (float*)take((size_t)NN * HEADS);
  float* sdst = (float*)take((size_t)NN * HEADS);
  unsigned* menc = (unsigned*)take((size_t)NN * HEADS);
  float* den = (float*)take((size_t)NN * HEADS);
  float* agg = (float*)take((size_t)NN * HID);
  float* logit = (float*)take((size_t)EE * HEADS);
  float* Ae = (float*)take(16);

  const int TB = 256;
  dim3 eg((EE + TB - 1) / TB), ng((NN + TB - 1) / TB),
      nh((NN * HID + TB - 1) / TB);
  // WPC waves per column tile; blocks = COLT*WPC/4 (128-thread blocks, 4 waves)
  constexpr int WPC = 1024;

  // ---- preprocess ----
  fix_edges<<<(EE * CE + TB - 1) / TB, TB, 0, stream>>>(EA, ea, EE * CE);
  gemm_node<CIN, HID, WPC, true, true, false, false><<<2 * WPC / 4, 128, 0, stream>>>(
      X, emb[0].W, emb[0].b, emb[0].gamma, emb[0].beta, emb[0].mean, emb[0].var,
      nullptr, xa, NN);
  gemm_node<HID, HID, WPC, true, false, true, true><<<2 * WPC / 4, 128, 0, stream>>>(
      xa, emb[1].W, emb[1].b, emb[1].gamma, emb[1].beta, emb[1].mean, emb[1].var,
      xa, xb, NN);

  float* xc = xb; float* xo = xa;
  float* ec = ea; float* eo = eb;
  for (int l = 0; l < NCONV; ++l) {
    const Conv& p = cv[l == 0 ? 0 : 1];  // conv0 then shared weights
    ae_precompute<<<1, 32, 0, stream>>>(p.We, p.a_e, Ae);
    gemm_node<HID, HID, WPC, false, false, false, false><<<2 * WPC / 4, 128, 0, stream>>>(
        xc, p.Wx, nullptr, nullptr, nullptr, nullptr, nullptr, nullptr, xp, NN);
    node_init<<<ng, TB, 0, stream>>>(xp, p.a_src, p.a_dst, ssrc, sdst, menc, den, agg);
    edge_logit<<<eg, TB, 0, stream>>>(EI, ec, ssrc, sdst, Ae, logit, menc);
    edge_exp<<<eg, TB, 0, stream>>>(EI, logit, menc, den);
    edge_agg<<<eg, TB, 0, stream>>>(EI, logit, menc, den, xp, agg);
    node_update<<<nh, TB, 0, stream>>>(xc, agg, p.bx, xo);
    { float* t = xc; xc = xo; xo = t; }
    edge_mlp<<<eg, TB, 0, stream>>>(xc, ec, EI, p.Wem, p.bem, eo);
    { float* t = ec; ec = eo; eo = t; }
  }

  // ---- FC head ----
  float* fb0 = xp; float* fb1 = agg;  // free after convs, NN*HID >= NN*FCC
  gemm_node<HID, FCC, WPC, true, false, false, false><<<WPC / 4, 128, 0, stream>>>(
      xc, fc[0].W, fc[0].b, fc[0].gamma, fc[0].beta, fc[0].mean, fc[0].var,
      nullptr, fb0, NN);
  const float* fin = fb0;
  for (int i = 1; i < NUMFC; ++i) {
    float* outp = (i == NUMFC - 1) ? (float*)d_out : ((i & 1) ? fb1 : fb0);
    gemm_node<FCC, FCC, WPC, true, false, false, false><<<WPC / 4, 128, 0, stream>>>(
        fin, fc[i].W, fc[i].b, fc[i].gamma, fc[i].beta, fc[i].mean, fc[i].var,
        nullptr, outp, NN);
    fin = outp;
  }
}